// EntityClassify_22282290332037
// MI455X (gfx1250) — compile-verified
//
#include <hip/hip_runtime.h>

typedef __attribute__((ext_vector_type(2))) float v2f;
typedef __attribute__((ext_vector_type(8))) float v8f;

// ---------------------------------------------------------------------------
// Degree counting (per relation) and reciprocal-degree table
// ---------------------------------------------------------------------------
__global__ __launch_bounds__(256) void ec_count_deg(const int* __restrict__ dst,
                                                    int* __restrict__ deg,
                                                    int total, int Nn, int Eper) {
  int t = blockIdx.x * blockDim.x + threadIdx.x;
  if (t >= total) return;
  int r = t / Eper;
  atomicAdd(&deg[r * Nn + dst[t]], 1);
}

__global__ __launch_bounds__(256) void ec_inv_deg(int* __restrict__ deg,
                                                  float* __restrict__ invd,
                                                  int total) {
  int t = blockIdx.x * blockDim.x + threadIdx.x;
  if (t >= total) return;
  float d = (float)deg[t];
  invd[t] = 1.0f / fmaxf(d, 1.0f);   // norm='right': divide by max(deg,1)
}

// ---------------------------------------------------------------------------
// fp32 WMMA GEMM: C[M,NC] = A[M,K] @ B[K,NC] (+ bias), row-major.
// B is staged in LDS k-pair-interleaved: Bs[(k/2)*S + 2n + (k&1)], so a WMMA
// B fragment (rows k+2h, k+2h+1 at column n) is ONE aligned ds_load_b64 into
// a consecutive VGPR pair -- no 2addr loads, no mov marshaling.
// Pair-row stride S = 32 (mod 64) dwords => the two lane-halves (pair-rows p
// and p+1) hit disjoint 32-bank ranges of the 64 LDS banks.
// One wave computes a 16 x NC output strip with v_wmma_f32_16x16x4_f32.
// A-fragment (16x4, 2 VGPR): VGPR0 = K {0,2} by lane-half, VGPR1 = K {1,3}.
// C/D (16x16, 8 VGPR): row = v + 8*(lane>>4), col = lane&15.
// ---------------------------------------------------------------------------
template <int K, int NC>
__global__ __launch_bounds__(128) void ec_gemm_wmma(
    const float* __restrict__ A, const float* __restrict__ B,
    const float* __restrict__ bias, float* __restrict__ C, int mtiles) {
  constexpr int NT = NC / 16;
  constexpr int S = ((2 * NC) % 64 == 32) ? (2 * NC) : (2 * NC + 32);
  __shared__ float Bs[(K / 2) * S];

  // Cooperative stage of B[K,NC] into pair-interleaved LDS layout.
  for (int i = threadIdx.x; i < K * NC / 4; i += 128) {
    int k = (i * 4) / NC;
    int n = (i * 4) % NC;
    float4 v = *(const float4*)(B + i * 4);
    float* p = &Bs[(k >> 1) * S + 2 * n + (k & 1)];
    p[0] = v.x;
    p[2] = v.y;
    p[4] = v.z;
    p[6] = v.w;
  }
  __syncthreads();

  int wave = (blockIdx.x * blockDim.x + threadIdx.x) >> 5;  // wave-uniform
  if (wave >= mtiles) return;                               // whole-wave exit
  int lane = threadIdx.x & 31;
  int m = lane & 15;
  int half = lane >> 4;

  const float* Arow = A + (size_t)(wave * 16 + m) * K + 2 * half;

  v8f acc[NT];
  v8f z = {0.f, 0.f, 0.f, 0.f, 0.f, 0.f, 0.f, 0.f};
#pragma unroll
  for (int j = 0; j < NT; ++j) acc[j] = z;

#pragma unroll 8
  for (int k = 0; k < K; k += 4) {
    float2 av = *(const float2*)(Arow + k);    // streaming A, 8B per lane
    v2f a = {av.x, av.y};
    // pair-row index of rows (k+2h, k+2h+1); column m of tile j at +32j dwords
    const float* Bp = &Bs[(size_t)((k + 2 * half) >> 1) * S + 2 * m];
#pragma unroll
    for (int j = 0; j < NT; ++j) {
      v2f b = *(const v2f*)(Bp + j * 32);      // one ds_load_b64
      acc[j] = __builtin_amdgcn_wmma_f32_16x16x4_f32(
          /*neg_a=*/false, a, /*neg_b=*/false, b,
          /*c_mod=*/(short)0, acc[j], /*reuse_a=*/false, /*reuse_b=*/false);
    }
  }

#pragma unroll
  for (int j = 0; j < NT; ++j) {
    float bv = bias ? bias[j * 16 + m] : 0.0f;
#pragma unroll
    for (int v = 0; v < 8; ++v) {
      C[(size_t)(wave * 16 + v + 8 * half) * NC + j * 16 + m] = acc[j][v] + bv;
    }
  }
}

// ---------------------------------------------------------------------------
// Edge aggregation: acc[dst] += hw[src] * inv_deg[dst] (per relation).
// COLS/4 threads per edge, float4 (b128) gathers of contiguous rows,
// native global_atomic_add_f32 scatter (accumulator is L2-resident).
// ---------------------------------------------------------------------------
template <int COLS>
__global__ __launch_bounds__(256) void ec_agg(
    const float* __restrict__ hw, const int* __restrict__ src,
    const int* __restrict__ dst, const float* __restrict__ invd,
    float* __restrict__ acc, int E) {
  constexpr int TPE = COLS / 4;
  int t = blockIdx.x * blockDim.x + threadIdx.x;
  int e = t / TPE;
  if (e >= E) return;
  int c = (t % TPE) * 4;
  int s = src[e];
  int d = dst[e];
  float w = invd[d];
  float4 v = *(const float4*)(hw + (size_t)s * COLS + c);
  float* dp = acc + (size_t)d * COLS + c;
  unsafeAtomicAdd(dp + 0, v.x * w);
  unsafeAtomicAdd(dp + 1, v.y * w);
  unsafeAtomicAdd(dp + 2, v.z * w);
  unsafeAtomicAdd(dp + 3, v.w * w);
}

__global__ __launch_bounds__(256) void ec_relu(float* __restrict__ h, int total) {
  int t = blockIdx.x * blockDim.x + threadIdx.x;
  if (t >= total) return;
  h[t] = fmaxf(h[t], 0.0f);
}

// ---------------------------------------------------------------------------
// Orchestration
// ---------------------------------------------------------------------------
extern "C" void kernel_launch(void* const* d_in, const int* in_sizes, int n_in,
                              void* d_out, int out_size, void* d_ws, size_t ws_size,
                              hipStream_t stream) {
  const float* x   = (const float*)d_in[0];   // [N, 256]
  const int*   src = (const int*)d_in[1];     // [R, E]
  const int*   dst = (const int*)d_in[2];     // [R, E]
  const float* W1  = (const float*)d_in[3];   // [R, 256, 64]
  const float* L1w = (const float*)d_in[4];   // [256, 64]
  const float* L1b = (const float*)d_in[5];   // [64]
  const float* W2  = (const float*)d_in[6];   // [R, 64, 16]
  const float* L2w = (const float*)d_in[7];   // [64, 16]
  const float* L2b = (const float*)d_in[8];   // [16]
  float* out = (float*)d_out;                 // [N, 16]

  const int D_IN = 256, D_H = 64, D_OUT = 16;
  const int Nn = in_sizes[0] / D_IN;           // 100000 (divisible by 16)
  const int R  = in_sizes[3] / (D_IN * D_H);   // 4
  const int E  = in_sizes[1] / R;              // 800000

  // Workspace layout (~53 MB): [deg/inv R*N] [h N*64] [hw N*64]
  char* ws = (char*)d_ws;
  size_t off = 0;
  int*   deg  = (int*)(ws + off);
  float* invd = (float*)(ws + off);
  off += (size_t)R * Nn * sizeof(float);
  off = (off + 255) & ~(size_t)255;
  float* h = (float*)(ws + off);               // hidden / layer-1 accumulator
  off += (size_t)Nn * D_H * sizeof(float);
  off = (off + 255) & ~(size_t)255;
  float* hw = (float*)(ws + off);              // per-relation transform buffer

  // 1) degrees -> reciprocal degrees
  hipMemsetAsync(deg, 0, (size_t)R * Nn * sizeof(int), stream);
  {
    int total = R * E;
    ec_count_deg<<<(total + 255) / 256, 256, 0, stream>>>(dst, deg, total, Nn, E);
  }
  {
    int total = R * Nn;
    ec_inv_deg<<<(total + 255) / 256, 256, 0, stream>>>(deg, invd, total);
  }

  int mtiles = Nn / 16;
  int gemmBlocks = (mtiles * 32 + 127) / 128;

  // 2) Layer 1: h = x@L1w + L1b  (non-atomic init), then += relation messages
  ec_gemm_wmma<256, 64><<<gemmBlocks, 128, 0, stream>>>(x, L1w, L1b, h, mtiles);
  for (int r = 0; r < R; ++r) {
    ec_gemm_wmma<256, 64><<<gemmBlocks, 128, 0, stream>>>(
        x, W1 + (size_t)r * D_IN * D_H, nullptr, hw, mtiles);
    long long tot = (long long)E * (D_H / 4);
    ec_agg<64><<<(int)((tot + 255) / 256), 256, 0, stream>>>(
        hw, src + (size_t)r * E, dst + (size_t)r * E, invd + (size_t)r * Nn, h, E);
  }
  ec_relu<<<(Nn * D_H + 255) / 256, 256, 0, stream>>>(h, Nn * D_H);

  // 3) Layer 2: out = h@L2w + L2b (overwrites poisoned d_out), then += messages
  ec_gemm_wmma<64, 16><<<gemmBlocks, 128, 0, stream>>>(h, L2w, L2b, out, mtiles);
  for (int r = 0; r < R; ++r) {
    ec_gemm_wmma<64, 16><<<gemmBlocks, 128, 0, stream>>>(
        h, W2 + (size_t)r * D_H * D_OUT, nullptr, hw, mtiles);
    long long tot = (long long)E * (D_OUT / 4);
    ec_agg<16><<<(int)((tot + 255) / 256), 256, 0, stream>>>(
        hw, src + (size_t)r * E, dst + (size_t)r * E, invd + (size_t)r * Nn, out, E);
  }
}